// SparseAdaptiveGraph_5909875000341
// MI455X (gfx1250) — compile-verified
//
#include <hip/hip_runtime.h>

typedef float v2f __attribute__((ext_vector_type(2)));
typedef float v8f __attribute__((ext_vector_type(8)));

#define NN 8192
#define DD 64
#define KTOP 32

// ---------------------------------------------------------------------------
// Kernel 1: C = relu(A @ B) using V_WMMA_F32_16X16X4_F32 (fp32 exact path).
// A: [8192,64] row-major, B: [64,8192] row-major, C: [8192,8192].
// One 16-row block per blockIdx.x; 8 waves, each owns 16 column tiles.
// ---------------------------------------------------------------------------
__global__ __launch_bounds__(256) void gemm_relu_kernel(const float* __restrict__ A,
                                                        const float* __restrict__ B,
                                                        float* __restrict__ C) {
  const int lane = threadIdx.x & 31;
  const int wave = threadIdx.x >> 5;   // 0..7
  const int l15  = lane & 15;
  const int hh   = lane >> 4;          // 0: K pair {0,1}; 1: K pair {2,3}
  const int rowBase = blockIdx.x * 16;

  // A fragment, 16 K-steps of 16x4 fp32.
  // Layout: lanes 0-15 hold M=lane, VGPR0=K0, VGPR1=K1; lanes 16-31 hold K2/K3.
  v2f afrag[16];
  const float* aptr = A + (size_t)(rowBase + l15) * DD + 2 * hh;
#pragma unroll
  for (int s = 0; s < 16; ++s)
    afrag[s] = *(const v2f*)(aptr + 4 * s);

  const int tile0 = blockIdx.y * 128 + wave;  // grid.y=4 -> 128 tiles per group
  for (int t = 0; t < 16; ++t) {
    const int colBase = (tile0 + t * 8) * 16;
    v8f c = {};
#pragma unroll
    for (int s = 0; s < 16; ++s) {
      // B fragment 4x16: lanes 0-15 col=lane K0(v0)/K1(v1); lanes 16-31 K2/K3.
      const float* bptr = B + (size_t)(4 * s + 2 * hh) * NN + colBase + l15;
      v2f b;
      b.x = bptr[0];
      b.y = bptr[NN];
      c = __builtin_amdgcn_wmma_f32_16x16x4_f32(false, afrag[s], false, b,
                                                (short)0, c, false, false);
    }
    // C layout: VGPR r -> row (r + 8*hh), col = l15.
    float* cp = C + (size_t)(rowBase + 8 * hh) * NN + colBase + l15;
#pragma unroll
    for (int r = 0; r < 8; ++r) {
      float v = c[r];
      cp[(size_t)r * NN] = v > 0.0f ? v : 0.0f;
    }
  }
}

// ---------------------------------------------------------------------------
// Block reductions (256 threads = 8 wave32).
// ---------------------------------------------------------------------------
__device__ __forceinline__ float blockSumF(float val, volatile float* smem) {
#pragma unroll
  for (int off = 16; off > 0; off >>= 1) val += __shfl_down(val, off, 32);
  const int lane = threadIdx.x & 31, w = threadIdx.x >> 5;
  __syncthreads();
  if (lane == 0) smem[w] = val;
  __syncthreads();
  if (threadIdx.x == 0) {
    float s = 0.0f;
    for (int i = 0; i < 8; ++i) s += smem[i];
    smem[8] = s;
  }
  __syncthreads();
  return smem[8];
}

__device__ __forceinline__ int blockSumI(int val, volatile int* smem) {
#pragma unroll
  for (int off = 16; off > 0; off >>= 1) val += __shfl_down(val, off, 32);
  const int lane = threadIdx.x & 31, w = threadIdx.x >> 5;
  __syncthreads();
  if (lane == 0) smem[w] = val;
  __syncthreads();
  if (threadIdx.x == 0) {
    int s = 0;
    for (int i = 0; i < 8; ++i) s += smem[i];
    smem[8] = s;
  }
  __syncthreads();
  return smem[8];
}

__device__ __forceinline__ float blockMaxF(float val, volatile float* smem) {
#pragma unroll
  for (int off = 16; off > 0; off >>= 1) val = fmaxf(val, __shfl_down(val, off, 32));
  const int lane = threadIdx.x & 31, w = threadIdx.x >> 5;
  __syncthreads();
  if (lane == 0) smem[w] = val;
  __syncthreads();
  if (threadIdx.x == 0) {
    float s = smem[0];
    for (int i = 1; i < 8; ++i) s = fmaxf(s, smem[i]);
    smem[8] = s;
  }
  __syncthreads();
  return smem[8];
}

// ---------------------------------------------------------------------------
// Kernel 2: per-row top-32 threshold (bitwise binary search, valid for v>=0)
// + full-row softmax, rewriting d_out in place. One block per row.
// ---------------------------------------------------------------------------
__global__ __launch_bounds__(256) void topk_softmax_kernel(float* __restrict__ M) {
  __shared__ float smemF[9];
  __shared__ int   smemI[9];
  __shared__ int   eqCols[64];
  __shared__ int   eqCnt;

  const int tid = threadIdx.x;
  float* rowp = M + (size_t)blockIdx.x * NN;

  float v[32];
#pragma unroll
  for (int j = 0; j < 32; ++j) v[j] = rowp[j * 256 + tid];

  // Row max (for numerically stable softmax, matching jax.nn.softmax).
  float lm = 0.0f;
#pragma unroll
  for (int j = 0; j < 32; ++j) lm = fmaxf(lm, v[j]);
  const float m = blockMaxF(lm, smemF);

  // Binary search on uint bit space: largest T with count(v >= T) >= 32.
  // All values are >= 0 post-relu, so float order == uint order.
  unsigned lo = 0u, hi = 0x7F800000u;
  while (hi - lo > 1u) {
    const unsigned mid = (lo + hi) >> 1;
    const float t = __uint_as_float(mid);
    int cnt = 0;
#pragma unroll
    for (int j = 0; j < 32; ++j) cnt += (v[j] >= t) ? 1 : 0;
    cnt = blockSumI(cnt, smemI);
    if (cnt >= KTOP) lo = mid; else hi = mid;
  }
  const float T = __uint_as_float(lo);  // the 32nd-largest value

  // Strictly-greater count and their exp-sum.
  int   lgt = 0;
  float lsg = 0.0f;
#pragma unroll
  for (int j = 0; j < 32; ++j) {
    if (v[j] > T) { lgt++; lsg += __expf(v[j] - m); }
  }
  const int   cgt = blockSumI(lgt, smemI);
  const float sgt = blockSumF(lsg, smemF);
  const int needed = KTOP - cgt;  // >= 1 by construction

  // Gather columns equal to T for tie-rank resolution (rare; benign if T==0).
  if (tid == 0) eqCnt = 0;
  __syncthreads();
#pragma unroll
  for (int j = 0; j < 32; ++j) {
    if (v[j] == T) {
      int p = atomicAdd(&eqCnt, 1);
      if (p < 64) eqCols[p] = j * 256 + tid;
    }
  }
  __syncthreads();
  const int ne = eqCnt < 64 ? eqCnt : 64;

  const float base = __expf(-m);
  const float eT   = __expf(T - m);
  // Exact Z regardless of which tied elements are kept:
  const float Z = sgt + (float)needed * eT + (float)(NN - KTOP) * base;
  const float invZ = 1.0f / Z;
  const float baseOut = base * invZ;

#pragma unroll
  for (int j = 0; j < 32; ++j) {
    const int col = j * 256 + tid;
    float out;
    if (v[j] > T) {
      out = __expf(v[j] - m) * invZ;
    } else if (v[j] == T) {
      int rank = 0;
      for (int q = 0; q < ne; ++q) rank += (eqCols[q] < col) ? 1 : 0;
      out = (rank < needed) ? eT * invZ : baseOut;
    } else {
      out = baseOut;
    }
    rowp[col] = out;
  }
}

// ---------------------------------------------------------------------------
extern "C" void kernel_launch(void* const* d_in, const int* in_sizes, int n_in,
                              void* d_out, int out_size, void* d_ws, size_t ws_size,
                              hipStream_t stream) {
  (void)in_sizes; (void)n_in; (void)out_size; (void)d_ws; (void)ws_size;
  const float* A = (const float*)d_in[0];   // nodevec1 [8192, 64]
  const float* B = (const float*)d_in[1];   // nodevec2 [64, 8192]
  float* out = (float*)d_out;               // [8192, 8192]

  dim3 g1(NN / 16, 4, 1);
  gemm_relu_kernel<<<g1, 256, 0, stream>>>(A, B, out);
  topk_softmax_kernel<<<NN, 256, 0, stream>>>(out);
}